// SelfKL_53927609369062
// MI455X (gfx1250) — compile-verified
//
#include <hip/hip_runtime.h>

typedef __attribute__((ext_vector_type(2))) float v2f;
typedef __attribute__((ext_vector_type(8))) float v8f;

#define B_ROWS   4096
#define NTILE    256            // 4096 / 16
#define EPSF     1e-4f
#define LOG_EPSF -9.210340371976182f   // ln(1e-4)

struct RowQ {
  float S0, S1, S2;
  float LS0, LS1, LS2;
  float T0, T1, T2;
  float LT0, LT1, LT2;
};

// Softmax / one-hot quantities for one row (C == 3). Branch-free (selects only)
// so EXEC stays all-ones around the WMMAs.
__device__ __forceinline__ RowQ compute_row(const float* __restrict__ src,
                                            const long long* __restrict__ tgt,
                                            int r) {
  float x0 = src[3 * r + 0];
  float x1 = src[3 * r + 1];
  float x2 = src[3 * r + 2];
  float mx = fmaxf(x0, fmaxf(x1, x2));
  float e0 = __expf(x0 - mx);
  float e1 = __expf(x1 - mx);
  float e2 = __expf(x2 - mx);
  float Z = e0 + e1 + e2;
  float invZ = 1.0f / Z;
  float lZ = __logf(Z);
  RowQ q;
  q.S0 = e0 * invZ;  q.S1 = e1 * invZ;  q.S2 = e2 * invZ;
  q.LS0 = x0 - mx - lZ;  q.LS1 = x1 - mx - lZ;  q.LS2 = x2 - mx - lZ;
  int t = (int)tgt[r];
  q.T0 = (t == 0) ? 1.0f : EPSF;
  q.T1 = (t == 1) ? 1.0f : EPSF;
  q.T2 = (t == 2) ? 1.0f : EPSF;
  q.LT0 = (t == 0) ? 0.0f : LOG_EPSF;
  q.LT1 = (t == 1) ? 0.0f : LOG_EPSF;
  q.LT2 = (t == 2) ? 0.0f : LOG_EPSF;
  return q;
}

__device__ __forceinline__ float wave_sum32(float v) {
#pragma unroll
  for (int off = 16; off > 0; off >>= 1)
    v += __shfl_xor(v, off, 32);
  return v;
}

// acc[0] = sum_i -T_i.LS_i   (loss_0 numerator)
// acc[1] = sum |u_i - M1_ij| (loss_1 numerator)
// acc[2] = sum |M2_ij|       (loss_2 numerator)
// acc[3] = sum |u_i - M3_ij| (loss_3 numerator)
__global__ void selfkl_init_kernel(double* __restrict__ acc) {
  if (threadIdx.x < 4) acc[threadIdx.x] = 0.0;
}

__global__ __launch_bounds__(256) void selfkl_loss0_kernel(
    const float* __restrict__ src, const long long* __restrict__ tgt,
    double* __restrict__ acc) {
  int i = blockIdx.x * blockDim.x + threadIdx.x;
  float l0 = 0.0f;
  if (i < B_ROWS) {
    RowQ q = compute_row(src, tgt, i);
    l0 = -(q.T0 * q.LS0 + q.T1 * q.LS1 + q.T2 * q.LS2);
  }
  float ws = wave_sum32(l0);
  if ((threadIdx.x & 31) == 0) atomicAdd(&acc[0], (double)ws);
}

// One wave per 1x8 strip of 16x16 tiles. 8192 strips total.
__global__ __launch_bounds__(256) void selfkl_pair_kernel(
    const float* __restrict__ src, const long long* __restrict__ tgt,
    double* __restrict__ acc) {
  const int lane = threadIdx.x & 31;
  const int wave = threadIdx.x >> 5;
  const int strip = blockIdx.x * 8 + wave;          // [0, 8192)
  const int tileI = strip >> 5;                     // [0, 256)
  const int tileJ0 = (strip & 31) << 3;             // [0, 256) step 8
  const int m = lane & 15;
  const int hi = lane >> 4;                         // 0: K0/K1  1: K2/K3

  // ---- A-side (row i) quantities, computed once per strip ----
  const int i = tileI * 16 + m;
  RowQ ri = compute_row(src, tgt, i);
  float sS = ri.S0 * ri.LS0 + ri.S1 * ri.LS1 + ri.S2 * ri.LS2;
  float sT = ri.T0 * ri.LT0 + ri.T1 * ri.LT1 + ri.T2 * ri.LT2;
  float u = sS - sT;

  // M1 A-rows: K = [S0,S1,S2,T0 | T1,T2,0,0]  (K=6 padded to 8, two WMMAs)
  v2f a1l, a1h, a2, a3;
  a1l.x = hi ? ri.S2 : ri.S0;                 // K2 : K0
  a1l.y = hi ? ri.T0 : ri.S1;                 // K3 : K1
  a1h.x = hi ? 0.0f : ri.T1;                  // K6 : K4
  a1h.y = hi ? 0.0f : ri.T2;                  // K7 : K5
  // M2 A-rows: (LT_i - LS_i) padded to K=4
  float d0 = ri.LT0 - ri.LS0, d1 = ri.LT1 - ri.LS1, d2 = ri.LT2 - ri.LS2;
  a2.x = hi ? d2 : d0;
  a2.y = hi ? 0.0f : d1;
  // M3 A-rows: (S_i - T_i) padded to K=4
  float g0 = ri.S0 - ri.T0, g1 = ri.S1 - ri.T1, g2 = ri.S2 - ri.T2;
  a3.x = hi ? g2 : g0;
  a3.y = hi ? 0.0f : g1;

  // u value for accumulator slot k: row (tileI*16 + 8*hi + k) -> lane 8*hi+k
  float uk[8];
#pragma unroll
  for (int k = 0; k < 8; ++k) uk[k] = __shfl(u, 8 * hi + k, 32);

  float s1 = 0.0f, s2 = 0.0f, s3 = 0.0f;

  for (int tj = 0; tj < 8; ++tj) {
    const int j = (tileJ0 + tj) * 16 + m;
    RowQ rj = compute_row(src, tgt, j);

    // M1 B-rows: K = [LS0,LS1,LS2,-LT0 | -LT1,-LT2,0,0]
    v2f b1l, b1h, b2, b3;
    b1l.x = hi ? rj.LS2 : rj.LS0;
    b1l.y = hi ? -rj.LT0 : rj.LS1;
    b1h.x = hi ? 0.0f : -rj.LT1;
    b1h.y = hi ? 0.0f : -rj.LT2;
    // M2 B-rows: T_j padded to K=4
    b2.x = hi ? rj.T2 : rj.T0;
    b2.y = hi ? 0.0f : rj.T1;
    // M3 B-rows: LS_j padded to K=4
    b3.x = hi ? rj.LS2 : rj.LS0;
    b3.y = hi ? 0.0f : rj.LS1;

    // M1 = S.LS^T - T.LT^T  (K=8 via two chained f32 WMMAs)
    v8f c1 = {};
    c1 = __builtin_amdgcn_wmma_f32_16x16x4_f32(false, a1l, false, b1l,
                                               (short)0, c1, false, false);
    c1 = __builtin_amdgcn_wmma_f32_16x16x4_f32(false, a1h, false, b1h,
                                               (short)0, c1, false, false);
    // M2 = (LT - LS).T^T
    v8f c2 = {};
    c2 = __builtin_amdgcn_wmma_f32_16x16x4_f32(false, a2, false, b2,
                                               (short)0, c2, false, false);
    // M3 = (S - T).LS^T
    v8f c3 = {};
    c3 = __builtin_amdgcn_wmma_f32_16x16x4_f32(false, a3, false, b3,
                                               (short)0, c3, false, false);

#pragma unroll
    for (int k = 0; k < 8; ++k) {
      s1 += fabsf(uk[k] - c1[k]);
      s2 += fabsf(c2[k]);
      s3 += fabsf(uk[k] - c3[k]);
    }
  }

  // wave reduce -> LDS block reduce -> one f64 atomic per loss per block
  s1 = wave_sum32(s1);
  s2 = wave_sum32(s2);
  s3 = wave_sum32(s3);

  __shared__ float red[8][3];
  if (lane == 0) {
    red[wave][0] = s1;
    red[wave][1] = s2;
    red[wave][2] = s3;
  }
  __syncthreads();
  if (threadIdx.x == 0) {
    float t1 = 0.0f, t2 = 0.0f, t3 = 0.0f;
#pragma unroll
    for (int w = 0; w < 8; ++w) {
      t1 += red[w][0];
      t2 += red[w][1];
      t3 += red[w][2];
    }
    atomicAdd(&acc[1], (double)t1);
    atomicAdd(&acc[2], (double)t2);
    atomicAdd(&acc[3], (double)t3);
  }
}

__global__ void selfkl_finalize_kernel(const double* __restrict__ acc,
                                       float* __restrict__ out) {
  if (threadIdx.x == 0 && blockIdx.x == 0) {
    const double n2 = (double)B_ROWS * (double)B_ROWS;
    double l0 = acc[0] / (double)B_ROWS;
    double l1 = (1e-4 + acc[1]) / n2;
    double l2 = (1e-4 + acc[2]) / n2;
    double l3 = (1e-4 + acc[3]) / n2;
    out[0] = (float)((l0 + l1) + (l2 + l3));
    out[1] = (float)l0;
    out[2] = (float)l1;
    out[3] = (float)l2;
    out[4] = (float)l3;
  }
}

extern "C" void kernel_launch(void* const* d_in, const int* in_sizes, int n_in,
                              void* d_out, int out_size, void* d_ws,
                              size_t ws_size, hipStream_t stream) {
  (void)in_sizes; (void)n_in; (void)out_size; (void)ws_size;
  const float* src = (const float*)d_in[0];          // [4096, 3] f32 logits
  const long long* tgt = (const long long*)d_in[1];  // [4096] int64 labels
  float* out = (float*)d_out;                        // 5 f32 outputs
  double* acc = (double*)d_ws;                       // 4 f64 accumulators

  selfkl_init_kernel<<<1, 32, 0, stream>>>(acc);
  selfkl_loss0_kernel<<<B_ROWS / 256, 256, 0, stream>>>(src, tgt, acc);
  selfkl_pair_kernel<<<(NTILE * 32) / 8, 256, 0, stream>>>(src, tgt, acc);
  selfkl_finalize_kernel<<<1, 32, 0, stream>>>(acc, out);
}